// FourierButterworthHFCFilter_59450937311646
// MI455X (gfx1250) — compile-verified
//
#include <hip/hip_runtime.h>
#include <math.h>

#define HW      512
#define N2      (HW * HW)          // 262144 elements per plane
#define NPLANE  48                 // B*C = 16*3
#define TOTAL   (NPLANE * N2)

typedef __attribute__((ext_vector_type(2))) float        v2f;
typedef __attribute__((ext_vector_type(8))) float        v8f;
typedef __attribute__((ext_vector_type(4))) unsigned int u32x4;
typedef __attribute__((ext_vector_type(4))) int          i32x4;
typedef __attribute__((ext_vector_type(8))) int          i32x8;

// ---------------------------------------------------------------------------
// D = A*B + C on the matrix core: 16x16 (f32) += 16x4 (f32) * 4x16 (f32)
// ---------------------------------------------------------------------------
__device__ __forceinline__ v8f wmma4(v2f a, v2f b, v8f c) {
  // 8 args: (neg_a, A, neg_b, B, c_mod, C, reuse_a, reuse_b)
  return __builtin_amdgcn_wmma_f32_16x16x4_f32(false, a, false, b, (short)0, c,
                                               false, false);
}

// ---------------------------------------------------------------------------
// TDM: async DMA of a (KSLAB rows x 32 cols) f32 tile of a row-major
// 512x512 tensor into LDS.  D# layout per CDNA5 ISA ch.8:
//   group0: [1:0]=count=1, [63:32]=lds byte addr, [120:64]=global byte addr,
//           [127:126]=type=2
//   group1: data_size=4B, tensor_dim0/1=512, tile_dim0=32, tile_dim1=KSLAB,
//           tensor_dim0_stride=512; groups 2/3 zero (2-D tile)
// Completion tracked by TENSORcnt (in-order per wave).
// ---------------------------------------------------------------------------
template <int KSLAB>
__device__ __forceinline__ void tdm_load_tile(const float* gptr, unsigned lds_off) {
  unsigned long long ga = (unsigned long long)(uintptr_t)gptr;
  u32x4 g0 = {1u,                                        // count = 1
              lds_off,                                   // LDS byte address
              (unsigned)(ga & 0xffffffffull),            // global addr [31:0]
              (unsigned)((ga >> 32) & 0x01ffffffu) | (2u << 30)};  // addr hi | type=2
  i32x8 g1 = {0x00020000,                                // data_size = 4 bytes
              (int)(512u << 16),                         // tensor_dim0 = 512 (lo16)
              (int)(512u << 16),                         // dim0 hi=0 | tensor_dim1 lo16
              (int)(32u << 16),                          // dim1 hi=0 | tile_dim0 = 32
              KSLAB,                                     // tile_dim1 | tile_dim2=0
              512,                                       // tensor_dim0_stride lo32
              0, 0};                                     // stride hi | dim1_stride
  i32x4 z4 = {0, 0, 0, 0};
#if defined(__clang_major__) && (__clang_major__ >= 23)
  i32x8 z8 = {0, 0, 0, 0, 0, 0, 0, 0};
  __builtin_amdgcn_tensor_load_to_lds(g0, g1, z4, z4, z8, 0);
#else
  __builtin_amdgcn_tensor_load_to_lds(g0, g1, z4, z4, 0);
#endif
}

// ---------------------------------------------------------------------------
// 1) DFT twiddle tables: COS[k*512+n] = cos(2*pi*k*n/512), SIN likewise.
//    Forward F = COS - i*SIN, inverse Finv = (COS + i*SIN)/512 (scale folded
//    into the abs kernel as 1/512^2). k*n mod 512 keeps args small & exact.
// ---------------------------------------------------------------------------
__global__ void twiddle_init_kernel(float* __restrict__ C, float* __restrict__ S) {
  int idx = blockIdx.x * blockDim.x + threadIdx.x;
  if (idx >= N2) return;
  int k = idx >> 9, n = idx & 511;
  int kn = (k * n) & 511;
  float th = (float)kn * (6.2831853071795864769f / 512.0f);
  float s, c;
  __sincosf(th, &s, &c);
  C[idx] = c;
  S[idx] = s;
}

// ---------------------------------------------------------------------------
// 2) Per-plane median (lower middle order stat) via 8192-bin LDS histogram
// ---------------------------------------------------------------------------
__global__ void median_kernel(const float* __restrict__ x, float* __restrict__ med) {
  __shared__ unsigned hist[8192];
  const int plane = blockIdx.x;
  const float* p = x + (size_t)plane * N2;
  for (int i = threadIdx.x; i < 8192; i += blockDim.x) hist[i] = 0u;
  __syncthreads();
  for (int i = threadIdx.x; i < N2; i += blockDim.x) {
    float v = p[i];
    int b = (int)floorf((v + 8.0f) * 512.0f);   // 8192 bins / 16.0 range
    b = b < 0 ? 0 : (b > 8191 ? 8191 : b);
    atomicAdd(&hist[b], 1u);
  }
  __syncthreads();
  if (threadIdx.x == 0) {
    const unsigned rank = (N2 - 1) / 2;         // 0-based lower-middle element
    unsigned cum = 0;
    int b = 0;
    for (; b < 8192; ++b) { cum += hist[b]; if (cum > rank) break; }
    float v = ((float)b + 0.5f) * (1.0f / 512.0f) - 8.0f;
    med[plane] = v + 0.2f;
  }
}

// ---------------------------------------------------------------------------
// 3) xp = mask*x + (1-mask)*median
// ---------------------------------------------------------------------------
__global__ void fill_kernel(const float* __restrict__ x, const float* __restrict__ mask,
                            const float* __restrict__ med, float* __restrict__ xp) {
  int i = blockIdx.x * blockDim.x + threadIdx.x;
  if (i >= TOTAL) return;
  int plane = i >> 18;                           // N2 == 2^18
  float m = mask[i];
  xp[i] = m * x[i] + (1.0f - m) * med[plane];
}

// ---------------------------------------------------------------------------
// 4) Batched complex GEMM on the matrix core, compile-time specialized on
//    operand shape and twiddle sign:
//      (Dr + i*Di) = (Ar + i*sA*Ai) * (Br + i*sB*Bi),  sA/sB = +-1
//    One wave = one 32x32 complex tile (2x2 grid of 16x16 WMMA tiles,
//    8 v8f accumulators).  B operand K-slabs (KSLAB x 32) are DMA'd into a
//    double-buffered LDS tile by the Tensor Data Mover, overlapped with the
//    WMMA loop and synchronized with s_wait_tensorcnt; A fragments stream
//    from global (b64 per lane) with next-slab prefetch.
// ---------------------------------------------------------------------------
template <bool HAS_AI, bool HAS_BI, bool NEG_AI, bool NEG_BI>
__global__ void __launch_bounds__(32)
cgemm_wmma_kernel(const float* __restrict__ Ar, const float* __restrict__ Ai,
                  const float* __restrict__ Br, const float* __restrict__ Bi,
                  float* __restrict__ Dr, float* __restrict__ Di,
                  int strideA, int strideB) {
  constexpr int KSLAB  = 32;                     // K-rows per TDM slab
  constexpr int NSLAB  = HW / KSLAB;             // 16 slabs
  constexpr int SLABSZ = KSLAB * 32;             // floats per slab tile

  __shared__ float sBr[2 * SLABSZ];
  __shared__ float sBi[HAS_BI ? 2 * SLABSZ : 1];

  const int lane  = threadIdx.x;
  const int m0    = blockIdx.y << 5;             // 32-row tile
  const int n0    = blockIdx.x << 5;             // 32-col tile
  const int plane = blockIdx.z;

  const float* arp = Ar + (size_t)plane * (size_t)strideA;
  const float* brp = Br + (size_t)plane * (size_t)strideB;
  const float* aip = HAS_AI ? (Ai + (size_t)plane * (size_t)strideA) : nullptr;
  const float* bip = HAS_BI ? (Bi + (size_t)plane * (size_t)strideB) : nullptr;

  const int lm    = lane & 15;
  const int khalf = (lane >> 4) << 1;            // 0 or 2

  const size_t arow0 = (size_t)(m0 + lm) * HW;   // A rows for M-tile 0
  const size_t arow1 = arow0 + (size_t)16 * HW;  // A rows for M-tile 1
  const int    nc0   = n0 + lm;                  // D cols for N-tile 0
  const int    nc1   = nc0 + 16;                 // D cols for N-tile 1

  const v8f zero = {0.f, 0.f, 0.f, 0.f, 0.f, 0.f, 0.f, 0.f};
  v8f accr00 = zero, accr01 = zero, accr10 = zero, accr11 = zero;
  v8f acci00 = zero, acci01 = zero, acci10 = zero, acci11 = zero;

  // prologue: DMA slab 0 into LDS buffer 0
  tdm_load_tile<KSLAB>(brp + n0, (unsigned)(uintptr_t)&sBr[0]);
  if constexpr (HAS_BI)
    tdm_load_tile<KSLAB>(bip + n0, (unsigned)(uintptr_t)&sBi[0]);

  for (int s = 0; s < NSLAB; ++s) {
    const int cur = s & 1;
    const float* curBr = &sBr[cur * SLABSZ];
    const float* curBi = HAS_BI ? &sBi[cur * SLABSZ] : nullptr;

    if (s + 1 < NSLAB) {                         // DMA next slab, prefetch A
      const int nxt = cur ^ 1;
      const size_t koff = (size_t)(s + 1) * KSLAB;
      tdm_load_tile<KSLAB>(brp + koff * HW + n0,
                           (unsigned)(uintptr_t)&sBr[nxt * SLABSZ]);
      if constexpr (HAS_BI)
        tdm_load_tile<KSLAB>(bip + koff * HW + n0,
                             (unsigned)(uintptr_t)&sBi[nxt * SLABSZ]);
      __builtin_prefetch(arp + arow0 + koff, 0, 1);
      __builtin_prefetch(arp + arow1 + koff, 0, 1);
      if constexpr (HAS_AI) {
        __builtin_prefetch(aip + arow0 + koff, 0, 1);
        __builtin_prefetch(aip + arow1 + koff, 0, 1);
      }
      // newest (HAS_BI?2:1) ops may still be in flight; slab s is complete
      __builtin_amdgcn_s_wait_tensorcnt(HAS_BI ? 2 : 1);
    } else {
      __builtin_amdgcn_s_wait_tensorcnt(0);
    }

#pragma unroll
    for (int kk = 0; kk < KSLAB; kk += 4) {
      const int ka = s * KSLAB + kk + khalf;     // global K for A fragments
      const int kl = kk + khalf;                 // slab-local K for B tiles

      v2f far0 = *(const v2f*)(arp + arow0 + ka);
      v2f far1 = *(const v2f*)(arp + arow1 + ka);
      v2f fbr0, fbr1;
      fbr0.x = curBr[(kl + 0) * 32 + lm];
      fbr0.y = curBr[(kl + 1) * 32 + lm];
      fbr1.x = curBr[(kl + 0) * 32 + lm + 16];
      fbr1.y = curBr[(kl + 1) * 32 + lm + 16];

      accr00 = wmma4(far0, fbr0, accr00);
      accr01 = wmma4(far0, fbr1, accr01);
      accr10 = wmma4(far1, fbr0, accr10);
      accr11 = wmma4(far1, fbr1, accr11);

      if constexpr (HAS_AI && HAS_BI) {
        v2f fai0 = *(const v2f*)(aip + arow0 + ka);
        v2f fai1 = *(const v2f*)(aip + arow1 + ka);
        if constexpr (NEG_AI) { fai0 = -fai0; fai1 = -fai1; }
        v2f fbi0, fbi1;
        fbi0.x = curBi[(kl + 0) * 32 + lm];
        fbi0.y = curBi[(kl + 1) * 32 + lm];
        fbi1.x = curBi[(kl + 0) * 32 + lm + 16];
        fbi1.y = curBi[(kl + 1) * 32 + lm + 16];
        if constexpr (NEG_BI) { fbi0 = -fbi0; fbi1 = -fbi1; }
        v2f fan0 = -fai0, fan1 = -fai1;
        accr00 = wmma4(fan0, fbi0, accr00);      // - Ai*Bi
        accr01 = wmma4(fan0, fbi1, accr01);
        accr10 = wmma4(fan1, fbi0, accr10);
        accr11 = wmma4(fan1, fbi1, accr11);
        acci00 = wmma4(far0, fbi0, acci00);      //   Ar*Bi
        acci01 = wmma4(far0, fbi1, acci01);
        acci10 = wmma4(far1, fbi0, acci10);
        acci11 = wmma4(far1, fbi1, acci11);
        acci00 = wmma4(fai0, fbr0, acci00);      //   Ai*Br
        acci01 = wmma4(fai0, fbr1, acci01);
        acci10 = wmma4(fai1, fbr0, acci10);
        acci11 = wmma4(fai1, fbr1, acci11);
      } else if constexpr (HAS_AI) {             // B purely real
        v2f fai0 = *(const v2f*)(aip + arow0 + ka);
        v2f fai1 = *(const v2f*)(aip + arow1 + ka);
        if constexpr (NEG_AI) { fai0 = -fai0; fai1 = -fai1; }
        acci00 = wmma4(fai0, fbr0, acci00);
        acci01 = wmma4(fai0, fbr1, acci01);
        acci10 = wmma4(fai1, fbr0, acci10);
        acci11 = wmma4(fai1, fbr1, acci11);
      } else if constexpr (HAS_BI) {             // A purely real
        v2f fbi0, fbi1;
        fbi0.x = curBi[(kl + 0) * 32 + lm];
        fbi0.y = curBi[(kl + 1) * 32 + lm];
        fbi1.x = curBi[(kl + 0) * 32 + lm + 16];
        fbi1.y = curBi[(kl + 1) * 32 + lm + 16];
        if constexpr (NEG_BI) { fbi0 = -fbi0; fbi1 = -fbi1; }
        acci00 = wmma4(far0, fbi0, acci00);
        acci01 = wmma4(far0, fbi1, acci01);
        acci10 = wmma4(far1, fbi0, acci10);
        acci11 = wmma4(far1, fbi1, acci11);
      }
    }
  }

  float* drp = Dr + (size_t)plane * N2;
  float* dip = Di + (size_t)plane * N2;
  const int r0 = m0 + ((lane >> 4) << 3);        // +8 for upper lane half
#pragma unroll
  for (int r = 0; r < 8; ++r) {
    const size_t row0 = (size_t)(r0 + r) * HW;
    const size_t row1 = (size_t)(r0 + 16 + r) * HW;
    drp[row0 + nc0] = accr00[r];
    drp[row0 + nc1] = accr01[r];
    drp[row1 + nc0] = accr10[r];
    drp[row1 + nc1] = accr11[r];
    dip[row0 + nc0] = acci00[r];
    dip[row0 + nc1] = acci01[r];
    dip[row1 + nc0] = acci10[r];
    dip[row1 + nc1] = acci11[r];
  }
}

// ---------------------------------------------------------------------------
// 5) Frequency-domain filter: X *= filter_map (broadcast over planes)
// ---------------------------------------------------------------------------
__global__ void filter_mul_kernel(float* __restrict__ Xr, float* __restrict__ Xi,
                                  const float* __restrict__ fm) {
  int i = blockIdx.x * blockDim.x + threadIdx.x;
  if (i >= TOTAL) return;
  float f = fm[i & (N2 - 1)];
  Xr[i] *= f;
  Xi[i] *= f;
}

// ---------------------------------------------------------------------------
// 6) res = |ifft| : magnitude with the 1/(512*512) inverse-FFT scale folded in
// ---------------------------------------------------------------------------
__global__ void abs_kernel(const float* __restrict__ Yr, const float* __restrict__ Yi,
                           float* __restrict__ res) {
  int i = blockIdx.x * blockDim.x + threadIdx.x;
  if (i >= TOTAL) return;
  float a = Yr[i], b = Yi[i];
  res[i] = sqrtf(a * a + b * b) * (1.0f / (float)N2);
}

// ---------------------------------------------------------------------------
// 7) Exact 3% / 97% percentiles of trunc(res*256)/256 via a histogram whose
//    16384 bins coincide with the 1/256 quantization levels.
// ---------------------------------------------------------------------------
__global__ void percentile_kernel(const float* __restrict__ res,
                                  float* __restrict__ lo, float* __restrict__ hi) {
  __shared__ unsigned hist[16384];
  const int plane = blockIdx.x;
  const float* p = res + (size_t)plane * N2;
  for (int i = threadIdx.x; i < 16384; i += blockDim.x) hist[i] = 0u;
  __syncthreads();
  for (int i = threadIdx.x; i < N2; i += blockDim.x) {
    int b = (int)(p[i] * 256.0f);                 // res >= 0, trunc == floor
    b = b < 0 ? 0 : (b > 16383 ? 16383 : b);
    atomicAdd(&hist[b], 1u);
  }
  __syncthreads();
  if (threadIdx.x == 0) {
    const double n1  = (double)(N2 - 1);
    const double rlo = 0.03 * n1;
    const double rhi = 0.97 * n1;
    unsigned t[4] = {(unsigned)rlo, (unsigned)rlo + 1u,
                     (unsigned)rhi, (unsigned)rhi + 1u};
    float v[4] = {0.f, 0.f, 0.f, 0.f};
    unsigned cum = 0;
    int j = 0;
    for (int b = 0; b < 16384 && j < 4; ++b) {
      cum += hist[b];
      while (j < 4 && cum > t[j]) { v[j] = (float)b * (1.0f / 256.0f); ++j; }
    }
    lo[plane] = v[0] + (float)(rlo - (double)t[0]) * (v[1] - v[0]);
    hi[plane] = v[2] + (float)(rhi - (double)t[2]) * (v[3] - v[2]);
  }
}

// ---------------------------------------------------------------------------
// 8) out = (res - lo) / (hi - lo) * mask
// ---------------------------------------------------------------------------
__global__ void normalize_kernel(const float* __restrict__ res,
                                 const float* __restrict__ mask,
                                 const float* __restrict__ lo,
                                 const float* __restrict__ hi,
                                 float* __restrict__ out) {
  int i = blockIdx.x * blockDim.x + threadIdx.x;
  if (i >= TOTAL) return;
  int plane = i >> 18;
  float l = lo[plane], h = hi[plane];
  out[i] = (res[i] - l) / (h - l) * mask[i];
}

// ---------------------------------------------------------------------------
extern "C" void kernel_launch(void* const* d_in, const int* in_sizes, int n_in,
                              void* d_out, int out_size, void* d_ws, size_t ws_size,
                              hipStream_t stream) {
  (void)in_sizes; (void)n_in; (void)out_size; (void)ws_size;

  const float* x    = (const float*)d_in[0];
  const float* mask = (const float*)d_in[1];
  const float* fm   = (const float*)d_in[2];
  float* out        = (float*)d_out;

  char* ws = (char*)d_ws;
  const size_t MB = (size_t)1 << 20;
  float* COS = (float*)(ws + 0 * MB);    // 1 MB
  float* SIN = (float*)(ws + 1 * MB);    // 1 MB
  float* med = (float*)(ws + 2 * MB);    // 48 floats
  float* lo  = med + 64;
  float* hi  = med + 128;
  float* P0  = (float*)(ws +   4 * MB);  // xp               (48 MB)
  float* P1  = (float*)(ws +  52 * MB);  // T1r / T2r        (48 MB)
  float* P2  = (float*)(ws + 100 * MB);  // T1i / T2i        (48 MB)
  float* P3  = (float*)(ws + 148 * MB);  // Xr  / Yr         (48 MB)
  float* P4  = (float*)(ws + 196 * MB);  // Xi  / Yi         (48 MB)
  float* P5  = (float*)(ws + 244 * MB);  // res              (48 MB)

  const dim3 b256(256);
  const dim3 gElem((TOTAL + 255) / 256);
  const dim3 gInit((N2 + 255) / 256);
  const dim3 gGemm(HW / 32, HW / 32, NPLANE);   // 16 x 16 x 48 waves
  const dim3 bGemm(32);                         // one wave32 per block

  twiddle_init_kernel<<<gInit, b256, 0, stream>>>(COS, SIN);
  median_kernel<<<dim3(NPLANE), b256, 0, stream>>>(x, med);
  fill_kernel<<<gElem, b256, 0, stream>>>(x, mask, med, P0);

  // forward: X = F * xp * F      with F = COS - i*SIN
  // T1 = F * xp          : A complex (Ai = -SIN), B real
  cgemm_wmma_kernel<true, false, true, false>
      <<<gGemm, bGemm, 0, stream>>>(COS, SIN, P0, nullptr, P1, P2, 0, N2);
  // X  = T1 * F          : A complex, B complex (Bi = -SIN)
  cgemm_wmma_kernel<true, true, false, true>
      <<<gGemm, bGemm, 0, stream>>>(P1, P2, COS, SIN, P3, P4, N2, 0);

  filter_mul_kernel<<<gElem, b256, 0, stream>>>(P3, P4, fm);

  // inverse: Y = Finv * X' * Finv  with Finv = (COS + i*SIN)/512 (scale in abs)
  cgemm_wmma_kernel<true, true, false, false>
      <<<gGemm, bGemm, 0, stream>>>(COS, SIN, P3, P4, P1, P2, 0, N2);
  cgemm_wmma_kernel<true, true, false, false>
      <<<gGemm, bGemm, 0, stream>>>(P1, P2, COS, SIN, P3, P4, N2, 0);

  abs_kernel<<<gElem, b256, 0, stream>>>(P3, P4, P5);
  percentile_kernel<<<dim3(NPLANE), b256, 0, stream>>>(P5, lo, hi);
  normalize_kernel<<<gElem, b256, 0, stream>>>(P5, mask, lo, hi, out);
}